// GaussianPointCloudRasterisation_71897752535540
// MI455X (gfx1250) — compile-verified
//
#include <hip/hip_runtime.h>
#include <cstddef>

typedef __attribute__((ext_vector_type(2))) float v2f;
typedef __attribute__((ext_vector_type(8))) float v8f;

#define TILE_W   16
#define T_W      32          // tiles per row (512/16)
#define N_TILES  1024
#define MG       128         // padded gaussians per tile
#define QSTRIDE  17          // LDS padding: 17 coprime with 64 banks -> conflict free

__launch_bounds__(256, 2)
__global__ void gsplat_raster_wmma(const float* __restrict__ tp_uv,
                                   const float* __restrict__ tp_depth,
                                   const float* __restrict__ tp_cov,
                                   const float* __restrict__ tp_alpha,
                                   const float* __restrict__ tp_color,
                                   const int*   __restrict__ tp_valid,
                                   float* __restrict__ out_img,
                                   float* __restrict__ out_dep,
                                   float* __restrict__ out_acc)
{
    __shared__ float s_i00[MG], s_i01[MG], s_i11[MG];
    __shared__ float s_cu[MG],  s_cv[MG],  s_nrm[MG];
    __shared__ float s_alp[MG], s_dep[MG];
    __shared__ float s_cr[MG],  s_cg[MG],  s_cb[MG];
    __shared__ int   s_val[MG];
    __shared__ float s_q[256 * QSTRIDE];

    const int t    = blockIdx.x;
    const int tid  = threadIdx.x;
    const int lane = tid & 31;
    const int wave = tid >> 5;
    const int tu   = t & (T_W - 1);
    const int tv   = t >> 5;

    // Prefetch next tile's gaussian payload into L2 (global_prefetch_b8).
    if (t + 1 < N_TILES) {
        __builtin_prefetch(tp_uv  + (size_t)(t + 1) * MG * 2 + (size_t)(tid & 127) * 2, 0, 1);
        __builtin_prefetch(tp_cov + (size_t)(t + 1) * MG * 4 + (size_t)(tid & 127) * 4, 0, 1);
    }

    // ---- Stage gaussian params to LDS (SoA), precompute inverse-cov coeffs ----
    if (tid < MG) {
        const int g = tid;
        const size_t base = (size_t)t * MG + g;
        // tile-local center coordinates (keeps f32 quadratic-form well conditioned)
        const float u = tp_uv[base * 2 + 0] - (float)(tu * TILE_W);
        const float v = tp_uv[base * 2 + 1] - (float)(tv * TILE_W);
        const float4 c = ((const float4*)tp_cov)[base];   // c00 c01 c10 c11
        const float det  = c.x * c.w - c.y * c.z;
        const float rdet = 1.0f / det;
        s_i00[g] = c.w * rdet;
        s_i01[g] = -c.y * rdet;
        s_i11[g] = c.x * rdet;
        s_cu[g]  = u;
        s_cv[g]  = v;
        s_nrm[g] = 1.0f / (6.28318530717958647692f * __builtin_sqrtf(det));
        s_alp[g] = tp_alpha[base];
        s_dep[g] = tp_depth[base];
        s_cr[g]  = tp_color[base * 3 + 0];
        s_cg[g]  = tp_color[base * 3 + 1];
        s_cb[g]  = tp_color[base * 3 + 2];
        s_val[g] = tp_valid[base];
    }
    __syncthreads();   // params visible to all waves; after this, waves run decoupled

    // ---- Per-wave pixel ownership: pixels [wave*32, wave*32+32) of the tile ----
    // A-matrix (16x4 f32, M=pixels, K=features) per ISA layout:
    //   lanes 0-15 : M=lane,    VGPR0=K0, VGPR1=K1
    //   lanes16-31 : M=lane-16, VGPR0=K2, VGPR1=K3
    const int  nlo = lane & 15;
    const bool lo  = lane < 16;
    const float pxl = (float)nlo + 0.5f;            // local x of pixel M=nlo
    const float py0 = (float)(2 * wave) + 0.5f;     // group 0 row
    const float py1 = py0 + 1.0f;                   // group 1 row
    // features: f0=px^2 f1=px*py f2=py^2 f3=px | f4=py f5=1 f6=f7=0
    const v2f a0_g0 = { lo ? pxl * pxl : py0 * py0,  lo ? pxl * py0 : pxl };
    const v2f a0_g1 = { lo ? pxl * pxl : py1 * py1,  lo ? pxl * py1 : pxl };
    const v2f a1_g0 = { lo ? py0 : 0.0f,             lo ? 1.0f : 0.0f };
    const v2f a1_g1 = { lo ? py1 : 0.0f,             lo ? 1.0f : 0.0f };

    // per-pixel blend state (lane owns pixel p)
    const int p = wave * 32 + lane;
    float T_i = 1.0f, ar = 0.0f, ag = 0.0f, ab = 0.0f, ad = 0.0f, dn = 0.0f;
    bool  sat = false;

    for (int gb = 0; gb < MG / 16; ++gb) {
        // ---- B-matrix (4x16 f32, K=features, N=gaussians) ----
        //   lanes 0-15 : N=lane,    VGPR0=K0, VGPR1=K1
        //   lanes16-31 : N=lane-16, VGPR0=K2, VGPR1=K3
        const int g   = gb * 16 + nlo;
        const float i00 = s_i00[g], i01 = s_i01[g], i11 = s_i11[g];
        const float u = s_cu[g], v = s_cv[g];
        const float c0 = i00;
        const float c1 = 2.0f * i01;
        const float c2 = i11;
        const float c3 = -2.0f * (i00 * u + i01 * v);
        const float c4 = -2.0f * (i01 * u + i11 * v);
        const float c5 = i00 * u * u + 2.0f * i01 * u * v + i11 * v * v;
        const v2f b0 = { lo ? c0 : c2,  lo ? c1 : c3 };
        const v2f b1 = { lo ? c4 : 0.0f, lo ? c5 : 0.0f };

        // Q[pixel][gaussian] = A x B (+) A' x B'   (two K=4 passes cover 6 features)
        v8f q0 = {};
        q0 = __builtin_amdgcn_wmma_f32_16x16x4_f32(false, a0_g0, false, b0, (short)0, q0, false, false);
        q0 = __builtin_amdgcn_wmma_f32_16x16x4_f32(false, a1_g0, false, b1, (short)0, q0, false, false);
        v8f q1 = {};
        q1 = __builtin_amdgcn_wmma_f32_16x16x4_f32(false, a0_g1, false, b0, (short)0, q1, false, false);
        q1 = __builtin_amdgcn_wmma_f32_16x16x4_f32(false, a1_g1, false, b1, (short)0, q1, false, false);

        // ---- scatter D (wmma layout) -> pixel-major LDS ----
        // D: VGPR r, lanes0-15 = (M=r, N=lane); lanes16-31 = (M=r+8, N=lane-16)
        // Each wave touches ONLY its own 32*QSTRIDE slice of s_q, and LDS ops
        // from one wave are serviced in order -> no workgroup barrier needed;
        // the compiler's s_wait_dscnt covers the cross-lane exchange.
        const int mbase = lo ? 0 : 8;
        const int pix0  = wave * 32;
        #pragma unroll
        for (int r = 0; r < 8; ++r) {
            s_q[(pix0 + mbase + r)      * QSTRIDE + nlo] = q0[r];
            s_q[(pix0 + 16 + mbase + r) * QSTRIDE + nlo] = q1[r];
        }

        // ---- strictly ordered blend of these 16 gaussians for this lane's pixel ----
        #pragma unroll
        for (int j = 0; j < 16; ++j) {
            const int   gg   = gb * 16 + j;
            const float q    = s_q[p * QSTRIDE + j];
            // exp(-0.5*q) = exp2(q * (-0.5*log2(e))) -> v_exp_f32
            const float dens = __builtin_amdgcn_exp2f(q * -0.72134752044448170368f) * s_nrm[gg];
            const float a    = fminf(dens * s_alp[gg], 0.99f);
            const bool  big  = a >= (1.0f / 255.0f);
            const bool  wsat = (1.0f - T_i * (1.0f - a)) >= 0.9999f;
            const bool  vm   = s_val[gg] > 0;
            const bool  ctr  = vm && big && (!sat) && (!wsat);
            const float aeff = ctr ? a : 0.0f;
            const float w    = aeff * T_i;
            ar += w * s_cr[gg];
            ag += w * s_cg[gg];
            ab += w * s_cb[gg];
            ad += w * s_dep[gg];
            dn += w;
            T_i *= (1.0f - aeff);
            sat = sat || (vm && big && wsat);
        }
    }

    // ---- write outputs: image (H,W,3), depth (H,W), acc_alpha (H,W) ----
    const int lv  = p >> 4;
    const int lu  = p & 15;
    const int y   = tv * TILE_W + lv;
    const int x   = tu * TILE_W + lu;
    const int idx = y * 512 + x;
    out_img[idx * 3 + 0] = ar;
    out_img[idx * 3 + 1] = ag;
    out_img[idx * 3 + 2] = ab;
    out_dep[idx] = ad / fmaxf(dn, 1e-6f);
    out_acc[idx] = 1.0f - T_i;
}

extern "C" void kernel_launch(void* const* d_in, const int* in_sizes, int n_in,
                              void* d_out, int out_size, void* d_ws, size_t ws_size,
                              hipStream_t stream) {
    const float* tp_uv    = (const float*)d_in[0];
    const float* tp_depth = (const float*)d_in[1];
    const float* tp_cov   = (const float*)d_in[2];
    const float* tp_alpha = (const float*)d_in[3];
    const float* tp_color = (const float*)d_in[4];
    const int*   tp_valid = (const int*)d_in[5];

    float* out     = (float*)d_out;
    float* out_img = out;                    // 512*512*3
    float* out_dep = out + 512 * 512 * 3;    // 512*512
    float* out_acc = out + 512 * 512 * 4;    // 512*512

    gsplat_raster_wmma<<<N_TILES, 256, 0, stream>>>(
        tp_uv, tp_depth, tp_cov, tp_alpha, tp_color, tp_valid,
        out_img, out_dep, out_acc);
}